// Sparsify1D_kactiveIOnline_23398981829300
// MI455X (gfx1250) — compile-verified
//
#include <hip/hip_runtime.h>
#include <stdint.h>

// Problem constants (from the reference).
#define ROWS   1024
#define COLS   102400
#define KSEL   25600
#define NT     1024                    // threads per block (32 waves, wave32)
#define NW     (NT / 32)               // 32 waves
#define DEPTH  3                       // async pipeline depth (LDS slots per wave)
#define TILES  (COLS / (NW * 128))     // 25 tiles per wave; tile = 32 lanes * 4 floats
#define HBINS  65536                   // 16-bit radix histogram (256 KB of LDS)
#define CHUNK  (HBINS / NT)            // 64 bins owned per thread in the scan

typedef __attribute__((ext_vector_type(4))) float    v4f;
typedef __attribute__((ext_vector_type(4))) int      v4i;
typedef __attribute__((ext_vector_type(4))) uint32_t v4u;
typedef __attribute__((address_space(1))) v4i        g_v4i;   // global int4
typedef __attribute__((address_space(3))) v4i        l_v4i;   // LDS int4

// ---------------------------------------------------------------------------
// Order-preserving float <-> uint32 key (monotonic: bigger float => bigger key)
// ---------------------------------------------------------------------------
__device__ __forceinline__ uint32_t fkey(float f) {
  uint32_t i = __float_as_uint(f);
  // positive: i | 0x80000000 ; negative: ~i
  return i ^ (0x80000000u | (uint32_t)(((int32_t)i) >> 31));
}
__device__ __forceinline__ float funkey(uint32_t o) {
  uint32_t i = (o & 0x80000000u) ? (o ^ 0x80000000u) : ~o;
  return __uint_as_float(i);
}

// ---------------------------------------------------------------------------
// CDNA5 async global->LDS copy (ASYNCcnt path) + wait
// ---------------------------------------------------------------------------
#if __has_builtin(__builtin_amdgcn_global_load_async_to_lds_b128)
#define ASYNC_VIA_BUILTIN 1
#endif

__device__ __forceinline__ void async_cp16(const float* g, float* l) {
#ifdef ASYNC_VIA_BUILTIN
  __builtin_amdgcn_global_load_async_to_lds_b128(
      (g_v4i*)g,   // AS1 int4*  (C-style cast drops const, adds addrspace)
      (l_v4i*)l,   // AS3 int4*
      0, 0);
#else
  uint32_t loff = (uint32_t)(uintptr_t)l;   // LDS byte address for VDST
  asm volatile("global_load_async_to_lds_b128 %0, %1, off"
               :: "v"(loff), "v"(g)
               : "memory");
#endif
}

#if __has_builtin(__builtin_amdgcn_s_wait_asynccnt)
#define WAIT_ASYNC(n) __builtin_amdgcn_s_wait_asynccnt(n)
#else
#define WAIT_ASYNC(n) asm volatile("s_wait_asynccnt %0" :: "i"(n) : "memory")
#endif

// ---------------------------------------------------------------------------
// Per-element histogram update
// ---------------------------------------------------------------------------
template<int PASS>
__device__ __forceinline__ void hist4(v4f v, uint32_t* hist, uint32_t binHi) {
#pragma unroll
  for (int c = 0; c < 4; ++c) {
    uint32_t u = fkey(v[c]);
    if (PASS == 0) {
      atomicAdd(&hist[u >> 16], 1u);                       // ds_add_u32
    } else {
      if ((u >> 16) == binHi) atomicAdd(&hist[u & 0xFFFFu], 1u);
    }
  }
}

// ---------------------------------------------------------------------------
// Stream one row-slice per wave through LDS with a DEPTH-deep async pipeline.
// gl = this lane's first global element, stride 128 floats between tiles.
// ll = this lane's LDS slot-0 address, stride 128 floats between slots.
// Each wave consumes only data it loaded itself -> only s_wait_asynccnt needed.
// ---------------------------------------------------------------------------
template<int PASS>
__device__ __forceinline__ void stream_hist(const float* __restrict__ gl,
                                            float* __restrict__ ll,
                                            uint32_t* __restrict__ hist,
                                            uint32_t binHi) {
#pragma unroll
  for (int p = 0; p < DEPTH; ++p)
    async_cp16(gl + p * 128, ll + p * 128);

  int i = 0;
  for (; i + DEPTH <= TILES; ++i) {
    WAIT_ASYNC(DEPTH - 1);            // oldest in-flight tile has landed
    v4f v = *(const v4f*)(ll + (i % DEPTH) * 128);
    hist4<PASS>(v, hist, binHi);
    if (i + DEPTH < TILES)            // refill slot AFTER consuming it
      async_cp16(gl + (i + DEPTH) * 128, ll + ((i + DEPTH) % DEPTH) * 128);
  }
  WAIT_ASYNC(0);                      // drain tail
  for (; i < TILES; ++i) {
    v4f v = *(const v4f*)(ll + (i % DEPTH) * 128);
    hist4<PASS>(v, hist, binHi);
  }
}

// ---------------------------------------------------------------------------
// Zero the 64K-bin histogram with b128 stores (16 per thread instead of 64 b32)
// ---------------------------------------------------------------------------
__device__ __forceinline__ void clear_hist(uint32_t* hist, int tid) {
  v4u* h4 = (v4u*)hist;                       // HBINS/4 = 16384 uint4 slots
  const v4u z = {0u, 0u, 0u, 0u};
#pragma unroll
  for (int j = 0; j < (HBINS / 4) / NT; ++j)  // 16 iterations
    h4[j * NT + tid] = z;
}

// ---------------------------------------------------------------------------
// Find bin b s.t. (#elements in bins > b) < K <= (#elements in bins >= b).
// Returns {bin, count_strictly_above_bin}. Uses scan[NT] + misc[] in LDS.
// ---------------------------------------------------------------------------
__device__ __forceinline__ uint2 radix_select(uint32_t* hist, uint32_t* scan,
                                              uint32_t* misc, uint32_t K, int tid) {
  uint32_t s = 0;
  const int base = tid * CHUNK;
#pragma unroll 8
  for (int j = 0; j < CHUNK; ++j) s += hist[base + j];
  scan[tid] = s;
  __syncthreads();
  // Hillis-Steele inclusive SUFFIX sum over 1024 chunk sums (10 rounds).
  for (int off = 1; off < NT; off <<= 1) {
    uint32_t v = scan[tid];
    if (tid + off < NT) v += scan[tid + off];
    __syncthreads();
    scan[tid] = v;
    __syncthreads();
  }
  const uint32_t mine = scan[tid];
  const uint32_t nxt  = (tid + 1 < NT) ? scan[tid + 1] : 0u;
  if (mine >= K && nxt < K) { misc[0] = (uint32_t)tid; misc[1] = nxt; }
  __syncthreads();
  if (tid == 0) {
    const uint32_t c = misc[0];
    uint32_t run = misc[1];                 // count above chunk c
    uint32_t bin = c * CHUNK, above = run;
    for (int j = CHUNK - 1; j >= 0; --j) {
      const uint32_t b = c * CHUNK + j;
      const uint32_t cnt = hist[b];
      if (run + cnt >= K) { bin = b; above = run; break; }
      run += cnt;
    }
    misc[2] = bin; misc[3] = above;
  }
  __syncthreads();
  return make_uint2(misc[2], misc[3]);
}

// ---------------------------------------------------------------------------
// One workgroup per row. Dynamic LDS layout:
//   hist[65536] | scan[1024] | misc[16] | stage[32 waves * DEPTH * 128 floats]
//   = 256KB + 4KB + 64B + 48KB = 308.1 KB  (<= 320 KB WGP LDS on CDNA5)
// ---------------------------------------------------------------------------
extern "C" __global__ void __launch_bounds__(NT, 1)
topk_sparsify_kernel(const float* __restrict__ x, float* __restrict__ out) {
  extern __shared__ uint32_t smem[];
  uint32_t* hist  = smem;
  uint32_t* scan  = hist + HBINS;
  uint32_t* misc  = scan + NT;
  float*    stage = (float*)(misc + 16);

  const int tid  = threadIdx.x;
  const int w    = tid >> 5;
  const int lane = tid & 31;
  const int row  = blockIdx.x;

  const float* rowp = x + (size_t)row * COLS;
  const float* gl   = rowp + w * (COLS / NW) + lane * 4;    // lane's stream base
  float*       ll   = stage + (w * DEPTH) * 128 + lane * 4; // lane's LDS slot 0

  // ---- Pass A: histogram of top 16 key bits ----
  clear_hist(hist, tid);
  __syncthreads();

  stream_hist<0>(gl, ll, hist, 0u);
  __syncthreads();

  uint2 selA = radix_select(hist, scan, misc, KSEL, tid);
  const uint32_t binHi = selA.x;
  const uint32_t k2    = KSEL - selA.y;    // still needed inside bucket binHi
  __syncthreads();

  // ---- Pass B: histogram of low 16 key bits within bucket binHi ----
  clear_hist(hist, tid);
  __syncthreads();

  stream_hist<1>(gl, ll, hist, binHi);
  __syncthreads();

  uint2 selB = radix_select(hist, scan, misc, k2, tid);
  const uint32_t tkey = (binHi << 16) | selB.x;   // exact k-th largest value
  const float tf = funkey(tkey);

  // ---- Pass C: masked write, exactly as reference: comp * x ----
  float* orow = out + (size_t)row * COLS;
#pragma unroll 5
  for (int i = 0; i < COLS / (NT * 4); ++i) {
    const v4f v = *(const v4f*)(rowp + i * (NT * 4) + tid * 4);
    v4f r;
#pragma unroll
    for (int c = 0; c < 4; ++c) {
      const float comp = (v[c] >= tf) ? 1.0f : 0.0f;
      r[c] = comp * v[c];
    }
    __builtin_nontemporal_store(r, (v4f*)(orow + i * (NT * 4) + tid * 4));
  }
}

// ---------------------------------------------------------------------------
extern "C" void kernel_launch(void* const* d_in, const int* in_sizes, int n_in,
                              void* d_out, int out_size, void* d_ws, size_t ws_size,
                              hipStream_t stream) {
  (void)in_sizes; (void)n_in; (void)d_ws; (void)ws_size; (void)out_size;
  const float* x = (const float*)d_in[0];
  float* out = (float*)d_out;
  const size_t shmem =
      (size_t)(HBINS + NT + 16 + NW * DEPTH * 128) * sizeof(uint32_t); // 315456 B
  // Dynamic LDS > 64 KB default cap: raise the attribute (non-enqueuing host
  // API; deterministic and graph-capture safe).
  (void)hipFuncSetAttribute((const void*)topk_sparsify_kernel,
                            hipFuncAttributeMaxDynamicSharedMemorySize,
                            (int)shmem);
  topk_sparsify_kernel<<<dim3(ROWS), dim3(NT), shmem, stream>>>(x, out);
}